// Emotion_Predictions_72121090834435
// MI455X (gfx1250) — compile-verified
//
#include <hip/hip_runtime.h>

typedef float v2f __attribute__((ext_vector_type(2)));
typedef float v4f __attribute__((ext_vector_type(4)));
typedef float v8f __attribute__((ext_vector_type(8)));
typedef unsigned int u32x4 __attribute__((ext_vector_type(4)));
typedef int i32x4 __attribute__((ext_vector_type(4)));
typedef int i32x8 __attribute__((ext_vector_type(8)));

static constexpr int B = 64, L = 1024, D = 512, TOPK = 32, WIN = 5;
static constexpr int NOE = L - TOPK; // 992

static constexpr size_t OFF_PRED  = 0;                                   // [B,L]
static constexpr size_t OFF_IDX   = OFF_PRED  + (size_t)B * L;           // [B,32]
static constexpr size_t OFF_PAIRE = OFF_IDX   + (size_t)B * TOPK;        // [B,160,2]
static constexpr size_t OFF_CAND  = OFF_PAIRE + (size_t)B * TOPK*WIN*2;  // [B,32,512]
static constexpr size_t OFF_CTXE  = OFF_CAND  + (size_t)B * TOPK*D;      // [B,32,5,512]
static constexpr size_t OFF_NOE   = OFF_CTXE  + (size_t)B * TOPK*WIN*D;  // [B,992,512]
static constexpr size_t OFF_CTXN  = OFF_NOE   + (size_t)B * NOE*D;       // [B,992,5,512]
static constexpr size_t OFF_PAIRN = OFF_CTXN  + (size_t)B * NOE*WIN*D;   // [B,4960,2]

// ---------------- K0: v = W_emo @ W_out ; s = b_emo . W_out + b_out ----------
__global__ __launch_bounds__(256)
void k_fuse(const float* __restrict__ W_emo, const float* __restrict__ b_emo,
            const float* __restrict__ W_out, const float* __restrict__ b_out,
            float* __restrict__ vfuse) {
  int t = threadIdx.x;
  for (int d = t; d < D; d += 256) {
    float acc = 0.f;
    const float* wr = W_emo + (size_t)d * D;
    for (int e = 0; e < D; ++e) acc += wr[e] * W_out[e];
    vfuse[d] = acc;
  }
  __shared__ float red[256];
  float p = 0.f;
  for (int e = t; e < D; e += 256) p += b_emo[e] * W_out[e];
  red[t] = p;
  __syncthreads();
  for (int s = 128; s > 0; s >>= 1) {
    if (t < s) red[t] += red[t + s];
    __syncthreads();
  }
  if (t == 0) vfuse[D] = red[0] + b_out[0];
}

// ---------------- K1: pred_e[b,l] = doc[b,l,:] . v + s -----------------------
__global__ __launch_bounds__(256)
void k_pred(const float* __restrict__ doc, const float* __restrict__ vfuse,
            float* __restrict__ dout) {
  int wave = threadIdx.x >> 5, lane = threadIdx.x & 31;
  size_t row = (size_t)blockIdx.x * 8 + wave;
  const float4* dr = (const float4*)(doc + row * D);
  const float4* vr = (const float4*)vfuse;
  float sum = 0.f;
  for (int i = lane; i < D / 4; i += 32) {
    float4 x = dr[i], y = vr[i];
    sum += x.x * y.x + x.y * y.y + x.z * y.z + x.w * y.w;
  }
  for (int off = 16; off > 0; off >>= 1) sum += __shfl_down(sum, off, 32);
  if (lane == 0) dout[OFF_PRED + row] = sum + vfuse[D];
}

// ---------------- K2: top-k, compaction, pair tensors ------------------------
__global__ __launch_bounds__(1024)
void k_topk(const float* __restrict__ dout_pred, float* __restrict__ dout,
            int* __restrict__ order, int* __restrict__ dmap) {
  __shared__ float sv[L];
  __shared__ float rv[L];
  __shared__ int   ri[L];
  __shared__ int   ords[L];
  const int t = threadIdx.x, b = blockIdx.x;
  sv[t] = dout_pred[OFF_PRED + (size_t)b * L + t];
  int myrank = -1;
  __syncthreads();
  for (int k = 0; k < TOPK; ++k) {
    rv[t] = sv[t]; ri[t] = t;
    __syncthreads();
    for (int s = 512; s > 0; s >>= 1) {
      if (t < s) {
        float v1 = rv[t + s]; int i1 = ri[t + s];
        if (v1 > rv[t] || (v1 == rv[t] && i1 < ri[t])) { rv[t] = v1; ri[t] = i1; }
      }
      __syncthreads();
    }
    int idx = ri[0];
    if (t == idx) { myrank = k; sv[t] = -3.402823466e38f; }
    if (t == 0) ords[k] = idx;
    __syncthreads();
  }
  // compaction of non-selected indices (ascending) via Hillis-Steele scan
  int flag = (myrank < 0) ? 1 : 0;
  ri[t] = flag;
  __syncthreads();
  for (int off = 1; off < L; off <<= 1) {
    int add = (t >= off) ? ri[t - off] : 0;
    __syncthreads();
    ri[t] += add;
    __syncthreads();
  }
  int dm;
  if (flag) { int pos = ri[t] - 1; ords[TOPK + pos] = t; dm = TOPK + pos; }
  else dm = myrank;
  __syncthreads();
  order[(size_t)b * L + t] = ords[t];
  dmap[(size_t)b * L + t] = dm;
  if (t < TOPK) dout[OFF_IDX + (size_t)b * TOPK + t] = (float)ords[t];
  if (t < TOPK * WIN) {
    int k = t / WIN, w = t % WIN;
    int src = ords[k];
    int ctx = min(max(src + w - 2, 0), L - 1);
    dout[OFF_PAIRE + ((size_t)b * TOPK * WIN + t) * 2 + 0] = (float)src;
    dout[OFF_PAIRE + ((size_t)b * TOPK * WIN + t) * 2 + 1] = (float)ctx;
  }
  for (int p = t; p < NOE * WIN; p += 1024) {
    int j = p / WIN, w = p % WIN;
    int src = ords[TOPK + j];
    int ctx = min(max(src + w - 2, 0), L - 1);
    dout[OFF_PAIRN + ((size_t)b * NOE * WIN + p) * 2 + 0] = (float)src;
    dout[OFF_PAIRN + ((size_t)b * NOE * WIN + p) * 2 + 1] = (float)ctx;
  }
}

// ---------------- K3: fused WMMA GEMM (TDM-staged A) -------------------------
// Block = 64 rows x 128 cols x {emo, con}. A tile (64x512 fp32) is loaded by
// the Tensor Data Mover with LDS padding (2 DW per 256 DW -> row stride 516 DW,
// bank-conflict-free for the 16x4 A-fragment reads). Emo rows are scattered
// straight into their cand / no-emotion output slots (nontemporal stores);
// con rows go to the L2-resident workspace.
__global__ __launch_bounds__(256)
void k_gemm(const float* __restrict__ doc,
            const float* __restrict__ W_emo, const float* __restrict__ b_emo,
            const float* __restrict__ W_con, const float* __restrict__ b_con,
            const int* __restrict__ dmap,
            float* __restrict__ dout, float* __restrict__ conws) {
  constexpr int AS = 516;              // padded row stride in DWORDs
  __shared__ float As[64 * AS];        // 132 KB
  __shared__ float sbe[128], sbc[128];
  __shared__ unsigned long long sbase[64];
  const int tid = threadIdx.x;
  const int rg = blockIdx.x >> 2, cg = blockIdx.x & 3;
  const size_t row0 = (size_t)rg * 64;
  const int ncol0 = cg * 128;

  // --- TDM: async 2D tile load Global -> LDS (wave 0 issues, all waves sync)
  if (tid < 32) {
    unsigned long long ga = (unsigned long long)(size_t)(doc + row0 * D);
    unsigned int lds = (unsigned int)(size_t)&As[0];
    u32x4 g0 = { 1u,                                     // count=1, user mode
                 lds,                                    // lds_addr (bytes)
                 (unsigned int)ga,                       // global_addr[31:0]
                 (unsigned int)(ga >> 32) | 0x80000000u  // addr[56:32] | type=2
    };
    // data_size=4B(2), pad_enable, pad_interval=256DW(7), pad_amount=2DW(1)
    i32x8 g1 = { (int)((2u << 16) | (1u << 20) | (7u << 22) | (1u << 25)),
                 (int)(512u << 16),        // tensor_dim0 = 512 (bits 79:48)
                 0,                        // tensor_dim0 hi / tensor_dim1 lo
                 (int)(1u | (512u << 16)), // tensor_dim1 = 65536; tile_dim0 = 512
                 64,                       // tile_dim1 = 64 rows
                 512,                      // tensor_dim0_stride = 512
                 0, 0 };
    i32x4 gz4 = { 0, 0, 0, 0 };
    i32x8 gz8 = { 0, 0, 0, 0, 0, 0, 0, 0 };
    __builtin_amdgcn_tensor_load_to_lds(g0, g1, gz4, gz4, gz8, 0);
  }
  for (int i = tid; i < 128; i += 256) {
    sbe[i] = b_emo[ncol0 + i];
    sbc[i] = b_con[ncol0 + i];
  }
  if (tid < 64) {
    size_t g = row0 + tid;
    int bb = (int)(g >> 10);
    int slot = dmap[g];
    sbase[tid] = (slot < TOPK)
        ? OFF_CAND + ((size_t)bb * TOPK + slot) * D
        : OFF_NOE  + ((size_t)bb * NOE + (slot - TOPK)) * D;
  }
  if (tid < 32) __builtin_amdgcn_s_wait_tensorcnt(0);
  __syncthreads();

  const int wave = tid >> 5, lane = tid & 31;
  const int rt = wave >> 1, ch = wave & 1;      // row tile (0..3), col half (0..1)
  const int row = lane & 15, koff = (lane >> 4) * 2;
  const float* Arow = &As[(rt * 16 + row) * AS];
  const int nb = ncol0 + ch * 64 + (lane & 15); // base output column for B frags

  v8f accE[4] = {v8f{}, v8f{}, v8f{}, v8f{}};
  v8f accC[4] = {v8f{}, v8f{}, v8f{}, v8f{}};

#pragma unroll 1
  for (int half = 0; half < 2; ++half) {
    const float* Ar = Arow + half * 258;        // 256 DW + 2 DW mid-row pad
    const int kbase = half * 256;
    for (int k = 0; k < 256; k += 4) {
      v2f a;
      a.x = Ar[k + koff];
      a.y = Ar[k + koff + 1];
      const float* we0 = W_emo + (size_t)(kbase + k + koff) * D + nb;
      const float* wc0 = W_con + (size_t)(kbase + k + koff) * D + nb;
#pragma unroll
      for (int j = 0; j < 4; ++j) {
        v2f be; be.x = we0[j * 16]; be.y = we0[D + j * 16];
        accE[j] = __builtin_amdgcn_wmma_f32_16x16x4_f32(
            false, a, false, be, (short)0, accE[j], false, false);
        v2f bc; bc.x = wc0[j * 16]; bc.y = wc0[D + j * 16];
        accC[j] = __builtin_amdgcn_wmma_f32_16x16x4_f32(
            false, a, false, bc, (short)0, accC[j], false, false);
      }
    }
  }

#pragma unroll
  for (int j = 0; j < 4; ++j) {
    int nl = ch * 64 + j * 16 + (lane & 15);
    int n = ncol0 + nl;
    float be = sbe[nl], bc = sbc[nl];
#pragma unroll
    for (int v = 0; v < 8; ++v) {
      int M = rt * 16 + v + ((lane >> 4) << 3);
      __builtin_nontemporal_store(accE[j][v] + be, &dout[sbase[M] + n]);
      conws[(row0 + M) * D + n] = accC[j][v] + bc;
    }
  }
}

// ---------------- K4: windowed context gather from L2-resident con_rep -------
__global__ __launch_bounds__(256)
void k_ctx(const float* __restrict__ conws, const int* __restrict__ order,
           float* __restrict__ dout) {
  int R = blockIdx.x * 2 + (threadIdx.x >> 7);   // output row id, B*L*WIN total
  int lane = threadIdx.x & 127;
  int b = R / (L * WIN);
  int rem = R - b * (L * WIN);
  int s = rem / WIN, w = rem - s * WIN;
  int src = order[(size_t)b * L + s];
  int ctx = min(max(src + w - 2, 0), L - 1);
  v4f val = ((const v4f*)(conws + ((size_t)b * L + ctx) * D))[lane];
  size_t ob = (s < TOPK)
      ? OFF_CTXE + (((size_t)b * TOPK + s) * WIN + w) * D
      : OFF_CTXN + (((size_t)b * NOE + (s - TOPK)) * WIN + w) * D;
  __builtin_nontemporal_store(val, (v4f*)(dout + ob) + lane);
}

extern "C" void kernel_launch(void* const* d_in, const int* in_sizes, int n_in,
                              void* d_out, int out_size, void* d_ws, size_t ws_size,
                              hipStream_t stream) {
  const float* doc   = (const float*)d_in[0];
  const float* W_emo = (const float*)d_in[1];
  const float* b_emo = (const float*)d_in[2];
  const float* W_con = (const float*)d_in[3];
  const float* b_con = (const float*)d_in[4];
  const float* W_out = (const float*)d_in[5];
  const float* b_out = (const float*)d_in[6];
  float* dout = (float*)d_out;

  float* vfuse = (float*)d_ws;                                     // 513 floats
  int*   order = (int*)((char*)d_ws + 4096);                       // B*L ints
  int*   dmap  = (int*)((char*)d_ws + 4096 + (size_t)B * L * 4);   // B*L ints
  float* conws = (float*)((char*)d_ws + (1u << 20));               // B*L*D floats (128 MB)

  k_fuse<<<1, 256, 0, stream>>>(W_emo, b_emo, W_out, b_out, vfuse);
  k_pred<<<(B * L) / 8, 256, 0, stream>>>(doc, vfuse, dout);
  k_topk<<<B, 1024, 0, stream>>>(dout, dout, order, dmap);
  k_gemm<<<(B * L / 64) * 4, 256, 0, stream>>>(doc, W_emo, b_emo, W_con, b_con,
                                               dmap, dout, conws);
  k_ctx<<<(B * L * WIN) / 2, 256, 0, stream>>>(conws, order, dout);
}